// GIN_82222853914925
// MI455X (gfx1250) — compile-verified
//
#include <hip/hip_runtime.h>

// ---------------------------------------------------------------------------
// GIN forward on MI455X (gfx1250, wave32).
// - Edge scatter / pooling: float4 gathers + float atomics (bandwidth-bound).
// - GEMMs: V_WMMA_F32_16X16X4_F32. 256-thread blocks (8 waves); the 16xK
//   weight slice is staged in LDS once per block; each wave owns a 64x16 C
//   tile; K-loop is chunked (16 K values) with all loads issued before the
//   WMMA chain so waits amortize over 16 matrix ops.
// - BN stats: LDS ds_add_f32 per-block accumulation, then global atomics.
// ---------------------------------------------------------------------------

#define N_NODES_C  100000
#define N_EDGES_C  3200000
#define N_GRAPHS_C 512
#define DIMMAX     96
#define CLS_C      10
#define NEG_SLOPE  0.01f
#define BN_EPS_C   1e-5f
#define KU         4          // k-steps (of 4) per pipelined chunk -> 16 K values

typedef __attribute__((ext_vector_type(2))) float v2f;
typedef __attribute__((ext_vector_type(8))) float v8f;

// ------------------------------- utility kernels ---------------------------

__global__ void zero_f32(float* __restrict__ p, long n) {
    long i = (long)blockIdx.x * blockDim.x + threadIdx.x;
    if (i < n) p[i] = 0.f;
}

__global__ void init_features(const int* __restrict__ deg,
                              const int* __restrict__ lab,
                              float* __restrict__ x) {
    long i = (long)blockIdx.x * blockDim.x + threadIdx.x;
    if (i >= (long)N_NODES_C * 96) return;
    int col  = (int)(i % 96);
    int node = (int)(i / 96);
    float v;
    if (col < 64) v = (deg[node] == col)      ? 1.f : 0.f;
    else          v = (lab[node] == col - 64) ? 1.f : 0.f;
    x[i] = v;
}

// One thread per (edge, 4-feature group): float4 gather + 4 float atomics.
__global__ void scatter_edges4(const float* __restrict__ x,
                               const int* __restrict__ src,
                               const int* __restrict__ dst,
                               float* __restrict__ agg, int d) {
    int d4 = d >> 2;
    long i = (long)blockIdx.x * blockDim.x + threadIdx.x;
    long total = (long)N_EDGES_C * d4;
    if (i >= total) return;
    int g = (int)(i % d4);
    int e = (int)(i / d4);
    int s = src[e], t = dst[e];
    const float4 v = *(const float4*)(x + (long)s * d + 4 * g);
    float* base = agg + (long)t * d + 4 * g;
    atomicAdd(base + 0, v.x);
    atomicAdd(base + 1, v.y);
    atomicAdd(base + 2, v.z);
    atomicAdd(base + 3, v.w);
}

__global__ void combine_eps(const float* __restrict__ x,
                            const float* __restrict__ agg,
                            const float* __restrict__ eps,
                            float* __restrict__ h, long n) {
    long i = (long)blockIdx.x * blockDim.x + threadIdx.x;
    if (i < n) h[i] = (1.f + eps[0]) * x[i] + agg[i];
}

// ------------------------------- WMMA GEMM ---------------------------------
// Y[nrows, dout] = A[nrows, K] @ W[dout, K]^T + bias
// Block: 256 threads = 8 waves, fixed 16-col slice (blockIdx.y).
// The 16xK weight slice lives in LDS (<= 6KB).  Each wave computes a 64x16
// output tile (4 stacked 16x16 C tiles sharing each B fragment).
// K must be a multiple of 16, dout a multiple of 16.
// A-frag (16x4 f32): lanes 0-15 -> M=lane, K={0,1}; lanes 16-31 -> K={2,3}.
// B-frag (4x16 f32): lanes 0-15 -> N=lane, K={0,1}; lanes 16-31 -> K={2,3}.
// C/D: VGPR v -> M = v (lanes 0-15) or v+8 (lanes 16-31), N = lane%16.
__global__ __launch_bounds__(256)
void gemm_wmma_f32(const float* __restrict__ A, const float* __restrict__ W,
                   const float* __restrict__ bias, float* __restrict__ Y,
                   int nrows, int K, int dout) {
    __shared__ float ldsW[16 * DIMMAX];

    int tid = threadIdx.x;
    int n0  = blockIdx.y * 16;

    // cooperative stage of the 16 x K weight slice into LDS
    for (int i = tid; i < 16 * K; i += 256) {
        int nn = i / K, kk = i - nn * K;
        ldsW[i] = W[(long)(n0 + nn) * K + kk];
    }
    __syncthreads();

    int wave = tid >> 5;
    int lane = tid & 31;
    int r    = lane & 15;          // M (for A) / N (for B/C)
    int kh   = (lane >> 4) << 1;   // 0 or 2: K sub-offset for this half-wave
    int m0   = blockIdx.x * 512 + wave * 64;

    v8f c0 = {}, c1 = {}, c2 = {}, c3 = {};

    int rmax = nrows - 1;
    const float* a0p = A + (long)min(m0 +      r, rmax) * K;
    const float* a1p = A + (long)min(m0 + 16 + r, rmax) * K;
    const float* a2p = A + (long)min(m0 + 32 + r, rmax) * K;
    const float* a3p = A + (long)min(m0 + 48 + r, rmax) * K;
    const float* wp  = ldsW + r * K + kh;

    for (int k0 = 0; k0 < K; k0 += 4 * KU) {
        v2f bf[KU], a0[KU], a1[KU], a2[KU], a3[KU];
        // issue all loads of this chunk before any WMMA: waits amortize
#pragma unroll
        for (int u = 0; u < KU; ++u)
            bf[u] = *(const v2f*)(wp + k0 + 4 * u);           // ds_load_b64
#pragma unroll
        for (int u = 0; u < KU; ++u) {
            int k = k0 + 4 * u + kh;
            a0[u] = *(const v2f*)(a0p + k);                   // global b64
            a1[u] = *(const v2f*)(a1p + k);
            a2[u] = *(const v2f*)(a2p + k);
            a3[u] = *(const v2f*)(a3p + k);
        }
#pragma unroll
        for (int u = 0; u < KU; ++u) {
            c0 = __builtin_amdgcn_wmma_f32_16x16x4_f32(false, a0[u], false, bf[u],
                                                       (short)0, c0, false, false);
            c1 = __builtin_amdgcn_wmma_f32_16x16x4_f32(false, a1[u], false, bf[u],
                                                       (short)0, c1, false, false);
            c2 = __builtin_amdgcn_wmma_f32_16x16x4_f32(false, a2[u], false, bf[u],
                                                       (short)0, c2, false, false);
            c3 = __builtin_amdgcn_wmma_f32_16x16x4_f32(false, a3[u], false, bf[u],
                                                       (short)0, c3, false, false);
        }
    }

    int   hi  = (lane >> 4) << 3;   // 0 or 8: row offset of upper half-wave
    int   col = n0 + r;
    float bv  = bias[col];
#pragma unroll
    for (int v = 0; v < 8; ++v) {
        int row = m0 + v + hi;
        if (row < nrows)      Y[(long)row * dout + col] = c0[v] + bv;
        row = m0 + 16 + v + hi;
        if (row < nrows)      Y[(long)row * dout + col] = c1[v] + bv;
        row = m0 + 32 + v + hi;
        if (row < nrows)      Y[(long)row * dout + col] = c2[v] + bv;
        row = m0 + 48 + v + hi;
        if (row < nrows)      Y[(long)row * dout + col] = c3[v] + bv;
    }
}

// ------------------------------- BatchNorm ---------------------------------

__global__ void bn_stats(const float* __restrict__ Y, float* __restrict__ sum,
                         float* __restrict__ sumsq, int nrows, int d) {
    __shared__ float ssum[DIMMAX];
    __shared__ float ssq[DIMMAX];
    for (int i = threadIdx.x; i < d; i += blockDim.x) { ssum[i] = 0.f; ssq[i] = 0.f; }
    __syncthreads();
    const int ROWS = 64;
    long base  = (long)blockIdx.x * ROWS * d;
    long nelem = (long)nrows * d;
    int  total = ROWS * d;
    for (int i = threadIdx.x; i < total; i += blockDim.x) {
        long idx = base + i;
        if (idx < nelem) {
            float v   = Y[idx];
            int   col = i % d;              // base is a multiple of d
            atomicAdd(&ssum[col], v);       // ds_add_f32
            atomicAdd(&ssq[col], v * v);
        }
    }
    __syncthreads();
    for (int i = threadIdx.x; i < d; i += blockDim.x) {
        atomicAdd(&sum[i], ssum[i]);
        atomicAdd(&sumsq[i], ssq[i]);
    }
}

__global__ void bn_finalize(const float* __restrict__ sum,
                            const float* __restrict__ sumsq,
                            const float* __restrict__ gamma,
                            const float* __restrict__ beta,
                            float* __restrict__ scale, float* __restrict__ shift,
                            int d, float invN) {
    int i = blockIdx.x * blockDim.x + threadIdx.x;
    if (i >= d) return;
    float mu  = sum[i] * invN;
    float var = sumsq[i] * invN - mu * mu;
    float s   = rsqrtf(var + BN_EPS_C) * gamma[i];
    scale[i] = s;
    shift[i] = beta[i] - mu * s;
}

__global__ void bn_apply_lrelu(const float* __restrict__ Y,
                               const float* __restrict__ scale,
                               const float* __restrict__ shift,
                               float* __restrict__ out, long n, int d) {
    long i = (long)blockIdx.x * blockDim.x + threadIdx.x;
    if (i >= n) return;
    int col = (int)(i % d);
    float v = fmaf(Y[i], scale[col], shift[col]);
    out[i] = v >= 0.f ? v : NEG_SLOPE * v;
}

// ------------------------------- pooling / FC ------------------------------

__global__ void pool_nodes(const float* __restrict__ x,
                           const int* __restrict__ batch,
                           float* __restrict__ pooled) {
    long i = (long)blockIdx.x * blockDim.x + threadIdx.x;
    if (i >= (long)N_NODES_C * 64) return;
    int f    = (int)(i & 63);
    int node = (int)(i >> 6);
    atomicAdd(&pooled[(long)batch[node] * 64 + f], x[i]);
}

__global__ void lrelu_inplace(float* __restrict__ p, long n) {
    long i = (long)blockIdx.x * blockDim.x + threadIdx.x;
    if (i >= n) return;
    float v = p[i];
    p[i] = v >= 0.f ? v : NEG_SLOPE * v;
}

__global__ void fc_final(const float* __restrict__ H, const float* __restrict__ Wb,
                         const float* __restrict__ bb, float* __restrict__ out) {
    int i = blockIdx.x * blockDim.x + threadIdx.x;  // 512*10 outputs
    if (i >= N_GRAPHS_C * CLS_C) return;
    int c = i % CLS_C, g = i / CLS_C;
    const float* h = H  + (long)g * 64;
    const float* w = Wb + (long)c * 64;
    float acc = bb[c];
#pragma unroll
    for (int k = 0; k < 64; ++k) acc = fmaf(h[k], w[k], acc);
    out[i] = acc;
}

// ------------------------------- host side ---------------------------------

struct LayerP {
    const float *eps, *W1, *b1, *g1, *be1, *W2, *b2, *g2, *be2;
};

extern "C" void kernel_launch(void* const* d_in, const int* in_sizes, int n_in,
                              void* d_out, int out_size, void* d_ws, size_t ws_size,
                              hipStream_t stream) {
    (void)n_in; (void)out_size; (void)ws_size;

    // ---- identify graph tensors (insertion vs sorted dict order) ----
    const int *deg, *lab, *edge, *batch;
    if (in_sizes[1] == 2 * N_EDGES_C) {     // sorted: batch, edge_index, node_deg, node_lab
        batch = (const int*)d_in[0]; edge = (const int*)d_in[1];
        deg   = (const int*)d_in[2]; lab  = (const int*)d_in[3];
    } else {                                 // insertion: node_deg, node_lab, edge_index, batch
        deg   = (const int*)d_in[0]; lab  = (const int*)d_in[1];
        edge  = (const int*)d_in[2]; batch = (const int*)d_in[3];
    }
    const int* src = edge;
    const int* dst = edge + N_EDGES_C;

    // ---- identify params (handle insertion-order and jax sorted-key order) ----
    LayerP L[3];
    const float *Wa = nullptr, *ba = nullptr, *Wb = nullptr, *bb = nullptr;
    int p = 4;
    auto F = [&](int k) { return (const float*)d_in[k]; };
    auto parse_layer = [&](int li) {
        LayerP& l = L[li];
        if (in_sizes[p] == 1) {   // insertion: eps,W1,b1,g1,be1,W2,b2,g2,be2
            l.eps = F(p); l.W1 = F(p + 1); l.b1 = F(p + 2); l.g1 = F(p + 3);
            l.be1 = F(p + 4); l.W2 = F(p + 5); l.b2 = F(p + 6);
            l.g2 = F(p + 7); l.be2 = F(p + 8);
        } else {                  // sorted: W1,W2,b1,b2,be1,be2,eps,g1,g2
            l.W1 = F(p); l.W2 = F(p + 1); l.b1 = F(p + 2); l.b2 = F(p + 3);
            l.be1 = F(p + 4); l.be2 = F(p + 5); l.eps = F(p + 6);
            l.g1 = F(p + 7); l.g2 = F(p + 8);
        }
        p += 9;
    };
    auto parse_fc = [&]() {
        if (in_sizes[p + 1] == 640) {  // sorted: Wa,Wb,ba,bb
            Wa = F(p); Wb = F(p + 1); ba = F(p + 2); bb = F(p + 3);
        } else {                       // insertion: Wa,ba,Wb,bb
            Wa = F(p); ba = F(p + 1); Wb = F(p + 2); bb = F(p + 3);
        }
        p += 4;
    };
    if (in_sizes[4] == 4096) {  // fc block first (sorted top-level: fc < layers)
        parse_fc(); parse_layer(0); parse_layer(1); parse_layer(2);
    } else {                    // layers first
        parse_layer(0); parse_layer(1); parse_layer(2); parse_fc();
    }

    // ---- workspace layout (floats) ----
    float* ws   = (float*)d_ws;
    const long NB = (long)N_NODES_C * 96;
    float* bufX   = ws;            // current node features [N, d]
    float* bufA   = bufX + NB;     // agg / GEMM outputs    [N, d]
    float* bufH   = bufA + NB;     // combined H / H2       [N, d]
    float* pooled = bufH + NB;     // [512, 64]
    float* fcH    = pooled + (long)N_GRAPHS_C * 64;  // [512, 64]
    float* s_sum  = fcH + (long)N_GRAPHS_C * 64;     // [96]
    float* s_sq   = s_sum + 96;                      // [96]
    float* s_scl  = s_sq + 96;                       // [96]
    float* s_sft  = s_scl + 96;                      // [96]

    auto nb = [](long n) { return (unsigned)((n + 255) / 256); };
    const int T = 256;
    const float invN = 1.0f / (float)N_NODES_C;

    // ---- initial one-hot features [N, 96] ----
    init_features<<<nb((long)N_NODES_C * 96), T, 0, stream>>>(deg, lab, bufX);

    const int d1s[3] = {96, 64, 64};
    const int d2s[3] = {64, 64, 64};

    for (int l = 0; l < 3; ++l) {
        const int d1 = d1s[l], d2 = d2s[l];
        const long nd1 = (long)N_NODES_C * d1;
        const long nd2 = (long)N_NODES_C * d2;

        // agg = segment_sum(x[src], dst)
        zero_f32<<<nb(nd1), T, 0, stream>>>(bufA, nd1);
        scatter_edges4<<<nb((long)N_EDGES_C * (d1 / 4)), T, 0, stream>>>(
            bufX, src, dst, bufA, d1);

        // H = (1+eps)*x + agg
        combine_eps<<<nb(nd1), T, 0, stream>>>(bufX, bufA, L[l].eps, bufH, nd1);

        // Y1 = H @ W1^T + b1   (into bufA; agg is dead)
        gemm_wmma_f32<<<dim3((N_NODES_C + 511) / 512, d1 / 16), 256, 0, stream>>>(
            bufH, L[l].W1, L[l].b1, bufA, N_NODES_C, d1, d1);

        // H2 = lrelu(BN(Y1))   (into bufH)
        zero_f32<<<1, T, 0, stream>>>(s_sum, 192);
        bn_stats<<<(N_NODES_C + 63) / 64, T, 0, stream>>>(bufA, s_sum, s_sq, N_NODES_C, d1);
        bn_finalize<<<1, 128, 0, stream>>>(s_sum, s_sq, L[l].g1, L[l].be1, s_scl, s_sft, d1, invN);
        bn_apply_lrelu<<<nb(nd1), T, 0, stream>>>(bufA, s_scl, s_sft, bufH, nd1, d1);

        // Y2 = H2 @ W2^T + b2  (into bufA)
        gemm_wmma_f32<<<dim3((N_NODES_C + 511) / 512, d2 / 16), 256, 0, stream>>>(
            bufH, L[l].W2, L[l].b2, bufA, N_NODES_C, d1, d2);

        // x = lrelu(BN(Y2))    (into bufX)
        zero_f32<<<1, T, 0, stream>>>(s_sum, 192);
        bn_stats<<<(N_NODES_C + 63) / 64, T, 0, stream>>>(bufA, s_sum, s_sq, N_NODES_C, d2);
        bn_finalize<<<1, 128, 0, stream>>>(s_sum, s_sq, L[l].g2, L[l].be2, s_scl, s_sft, d2, invN);
        bn_apply_lrelu<<<nb(nd2), T, 0, stream>>>(bufA, s_scl, s_sft, bufX, nd2, d2);
    }

    // ---- global_add_pool over sorted graph ids ----
    const long npool = (long)N_GRAPHS_C * 64;
    zero_f32<<<nb(npool), T, 0, stream>>>(pooled, npool);
    pool_nodes<<<nb((long)N_NODES_C * 64), T, 0, stream>>>(bufX, batch, pooled);

    // ---- classifier head ----
    lrelu_inplace<<<nb(npool), T, 0, stream>>>(pooled, npool);
    gemm_wmma_f32<<<dim3(1, 64 / 16), 256, 0, stream>>>(
        pooled, Wa, ba, fcH, N_GRAPHS_C, 64, 64);
    fc_final<<<nb((long)N_GRAPHS_C * CLS_C), T, 0, stream>>>(fcH, Wb, bb, (float*)d_out);
}